// LGNN_39556648796683
// MI455X (gfx1250) — compile-verified
//
#include <hip/hip_runtime.h>
#include <hip/hip_bf16.h>

// ---------------------------------------------------------------------------
// LGNN for MI455X (gfx1250): fp32 WMMA (V_WMMA_F32_16X16X4_F32) GEMMs,
// LDS-staged operands with software-pipelined fragment fetch, atomic edge
// scatter, wave32 LayerNorm.
// ---------------------------------------------------------------------------

typedef float v2f __attribute__((ext_vector_type(2)));
typedef float v8f __attribute__((ext_vector_type(8)));

#define D 128

// ---------------- degree kernels ----------------
__global__ void deg_kernel(const int* __restrict__ idx, float* __restrict__ cnt, int E) {
  int t = blockIdx.x * blockDim.x + threadIdx.x;
  if (t < E) atomicAdd(&cnt[idx[t]], 1.0f);
}

__global__ void scale_kernel(float* __restrict__ cnt, int n) {
  int t = blockIdx.x * blockDim.x + threadIdx.x;
  if (t < n) cnt[t] = rsqrtf(cnt[t] + 1.0f);   // +1 self loop; always >= 1
}

// ---------------- gather: out[r] = src[idx[r]] ----------------
__global__ void gather_kernel(const float* __restrict__ src, const int* __restrict__ idx,
                              float* __restrict__ out, int nrows) {
  int t = blockIdx.x * blockDim.x + threadIdx.x;
  int r = t >> 5, c = (t & 31) << 2;
  if (r >= nrows) return;
  float4 v = *(const float4*)(src + (size_t)idx[r] * D + c);
  *(float4*)(out + (size_t)r * D + c) = v;
}

// ---------------- scatter-add over edges ----------------
// agg[didx[e]] += src[sidx ? sidx[e] : e]; 32 lanes x float4 per edge.
__global__ void scatter_kernel(const float* __restrict__ src, const int* __restrict__ sidx,
                               const int* __restrict__ didx, float* __restrict__ agg, int E) {
  int t = blockIdx.x * blockDim.x + threadIdx.x;
  int e = t >> 5, c = (t & 31) << 2;
  if (e >= E) return;
  int sr = sidx ? sidx[e] : e;
  int dr = didx[e];
  float4 v = *(const float4*)(src + (size_t)sr * D + c);
  float* d = agg + (size_t)dr * D + c;
  atomicAdd(d + 0, v.x);
  atomicAdd(d + 1, v.y);
  atomicAdd(d + 2, v.z);
  atomicAdd(d + 3, v.w);
}

// ---------------- gconv epilogue + relu-concat accumulation ----------------
// fs = ((agg + hn_selfloop) * rsqrt(deg_in) + bias) * sw
// resA (+)= relu(fs); resB (+)= fs
__global__ void fuse_kernel(const float* __restrict__ agg, const float* __restrict__ hn,
                            const float* __restrict__ sin_, const float* __restrict__ bias,
                            const float* __restrict__ sw, float* __restrict__ resA,
                            float* __restrict__ resB, int n, int init) {
  int t = blockIdx.x * blockDim.x + threadIdx.x;
  int r = t >> 5, c = (t & 31) << 2;
  if (r >= n) return;
  size_t off = (size_t)r * D + c;
  float4 a = *(const float4*)(agg + off);
  float4 h = *(const float4*)(hn + off);
  float  s = sin_[r];
  float4 bb = *(const float4*)(bias + c);
  float4 ww = *(const float4*)(sw + c);
  float f0 = ((a.x + h.x) * s + bb.x) * ww.x;
  float f1 = ((a.y + h.y) * s + bb.y) * ww.y;
  float f2 = ((a.z + h.z) * s + bb.z) * ww.z;
  float f3 = ((a.w + h.w) * s + bb.w) * ww.w;
  float4 rA = {fmaxf(f0, 0.f), fmaxf(f1, 0.f), fmaxf(f2, 0.f), fmaxf(f3, 0.f)};
  float4 rB = {f0, f1, f2, f3};
  if (!init) {
    float4 oA = *(const float4*)(resA + off);
    float4 oB = *(const float4*)(resB + off);
    rA.x += oA.x; rA.y += oA.y; rA.z += oA.z; rA.w += oA.w;
    rB.x += oB.x; rB.y += oB.y; rB.z += oB.z; rB.w += oB.w;
  }
  *(float4*)(resA + off) = rA;
  *(float4*)(resB + off) = rB;
}

// ---------------- WMMA GEMM: Y[n,128] = (X * rowscale?) @ W[K,128] ----------------
// K=128 (gconv, X=XA) or K=256 (cat, X=[XA|XB]).
// Block = (ROWS/16) waves; each wave -> 16 rows x 128 cols (8 WMMA tiles).
// W staged in LDS with K-pair interleave so B fragments are single b64 loads;
// X strip staged with +4 padding (conflict-free A column reads) and fused row
// scaling. Inner K loop is software-pipelined: next step's A + 8 B fragments
// are fetched while the current step's 8 WMMAs occupy the XDL pipe, so the
// LDS latency is hidden instead of a s_wait_dscnt 0 stall per WMMA pair.
template <int K, int ROWS>
__global__ void __launch_bounds__((ROWS / 16) * 32)
gemm_wmma(const float* __restrict__ XA, const float* __restrict__ XB,
          const float* __restrict__ rowscale, const float* __restrict__ W,
          float* __restrict__ Y, int n) {
  constexpr int WAVES = ROWS / 16;
  constexpr int XS = K + 4;                       // padded X stride
  extern __shared__ __align__(16) float smem[];
  float* Xs = smem;                               // ROWS * XS
  float* Ws = smem + ROWS * XS;                   // (K/2) * 256, pair-interleaved

  const int tid = threadIdx.x;
  const int nth = WAVES * 32;
  const int block_row = blockIdx.x * ROWS;

  // Stage W: element (r,c) -> Ws[(r/2)*256 + 2c + (r&1)]  (pairs contiguous per col)
  for (int i = tid * 4; i < K * D; i += nth * 4) {
    float4 w = *(const float4*)(W + i);
    int r = i >> 7, c = i & (D - 1);
    int base = (r >> 1) * 256 + (r & 1);
    Ws[base + 2 * (c + 0)] = w.x;
    Ws[base + 2 * (c + 1)] = w.y;
    Ws[base + 2 * (c + 2)] = w.z;
    Ws[base + 2 * (c + 3)] = w.w;
  }
  // Stage X strip with fused row scaling; zero-pad rows >= n
  for (int i = tid * 4; i < ROWS * K; i += nth * 4) {
    int r = i / K, c = i % K;
    int gr = block_row + r;
    float4 v = {0.f, 0.f, 0.f, 0.f};
    if (gr < n) {
      const float* s;
      if (K == 256) s = (c < D) ? (XA + (size_t)gr * D + c) : (XB + (size_t)gr * D + (c - D));
      else          s = XA + (size_t)gr * D + c;
      v = *(const float4*)s;
      if (rowscale) { float sc = rowscale[gr]; v.x *= sc; v.y *= sc; v.z *= sc; v.w *= sc; }
    }
    *(float4*)(Xs + r * XS + c) = v;
  }
  __syncthreads();

  const int wave = tid >> 5, lane = tid & 31;
  const int m = lane & 15, hi = lane >> 4;        // ISA 7.12.2 fragment addressing

  v8f acc[8];
#pragma unroll
  for (int t = 0; t < 8; ++t)
#pragma unroll
    for (int j = 0; j < 8; ++j) acc[t][j] = 0.0f;

  const float* xrow = Xs + (wave * 16 + m) * XS + 2 * hi;
  const int bcol = 2 * m;

  // prologue: fragments for k = 0
  v2f a_cur = *(const v2f*)(xrow);
  v2f b_cur[8];
  {
    const float* wrow = Ws + hi * 256;
#pragma unroll
    for (int t = 0; t < 8; ++t) b_cur[t] = *(const v2f*)(wrow + bcol + 32 * t);
  }

#pragma unroll 2
  for (int k = 0; k < K - 4; k += 4) {
    // prefetch k+4 fragments (issued before the WMMAs consume current ones)
    v2f a_nxt = *(const v2f*)(xrow + k + 4);
    v2f b_nxt[8];
    const float* wrow = Ws + ((k + 4) / 2 + hi) * 256;
#pragma unroll
    for (int t = 0; t < 8; ++t) b_nxt[t] = *(const v2f*)(wrow + bcol + 32 * t);

#pragma unroll
    for (int t = 0; t < 8; ++t)
      acc[t] = __builtin_amdgcn_wmma_f32_16x16x4_f32(
          false, a_cur, false, b_cur[t], (short)0, acc[t], false, false);

    a_cur = a_nxt;
#pragma unroll
    for (int t = 0; t < 8; ++t) b_cur[t] = b_nxt[t];
  }
  // epilogue: last k-step
#pragma unroll
  for (int t = 0; t < 8; ++t)
    acc[t] = __builtin_amdgcn_wmma_f32_16x16x4_f32(
        false, a_cur, false, b_cur[t], (short)0, acc[t], false, false);

  // C layout: VGPR j -> M = j + 8*hi, lane m -> N
  const int row0 = block_row + wave * 16 + hi * 8;
#pragma unroll
  for (int t = 0; t < 8; ++t) {
    int col = t * 16 + m;
#pragma unroll
    for (int j = 0; j < 8; ++j) {
      int r = row0 + j;
      if (r < n) Y[(size_t)r * D + col] = acc[t][j];
    }
  }
}

// ---------------- LayerNorm (one wave32 per row) ----------------
__global__ void ln_kernel(const float* __restrict__ x, const float* __restrict__ catb,
                          const float* __restrict__ g, const float* __restrict__ b,
                          float* __restrict__ y, int n) {
  int row = blockIdx.x * 8 + (threadIdx.x >> 5);
  int lane = threadIdx.x & 31;
  if (row >= n) return;
  int c = lane << 2;
  size_t off = (size_t)row * D + c;
  float4 v = *(const float4*)(x + off);
  float4 cb = *(const float4*)(catb + c);
  v.x += cb.x; v.y += cb.y; v.z += cb.z; v.w += cb.w;
  float s  = v.x + v.y + v.z + v.w;
  float sq = v.x * v.x + v.y * v.y + v.z * v.z + v.w * v.w;
#pragma unroll
  for (int o = 16; o > 0; o >>= 1) {
    s  += __shfl_xor(s, o, 32);
    sq += __shfl_xor(sq, o, 32);
  }
  float mu  = s * (1.0f / D);
  float var = sq * (1.0f / D) - mu * mu;
  float inv = rsqrtf(var + 1e-5f);
  float4 gg = *(const float4*)(g + c);
  float4 bb = *(const float4*)(b + c);
  float4 o4 = {(v.x - mu) * inv * gg.x + bb.x, (v.y - mu) * inv * gg.y + bb.y,
               (v.z - mu) * inv * gg.z + bb.z, (v.w - mu) * inv * gg.w + bb.w};
  *(float4*)(y + off) = o4;
}

// ---------------- host orchestration ----------------
static inline int cd(long long a, long long b) { return (int)((a + b - 1) / b); }

extern "C" void kernel_launch(void* const* d_in, const int* in_sizes, int n_in,
                              void* d_out, int out_size, void* d_ws, size_t ws_size,
                              hipStream_t stream) {
  const float* h0 = (const float*)d_in[0];
  const float* h1 = (const float*)d_in[1];
  const float* h2 = (const float*)d_in[2];
  const int* src0 = (const int*)d_in[3];
  const int* dst0 = (const int*)d_in[4];
  const int* src1 = (const int*)d_in[5];
  const int* dst1 = (const int*)d_in[6];
  const int* src2 = (const int*)d_in[7];
  const int* dst2 = (const int*)d_in[8];
  const float* convW = (const float*)d_in[9];
  const float* convb = (const float*)d_in[10];
  const float* fusW  = (const float*)d_in[11];
  const float* fusb  = (const float*)d_in[12];
  const float* catW  = (const float*)d_in[13];
  const float* catb  = (const float*)d_in[14];
  const float* convw = (const float*)d_in[15];
  const float* tdw   = (const float*)d_in[16];
  const float* buw   = (const float*)d_in[17];
  const float* lng   = (const float*)d_in[18];
  const float* lnb   = (const float*)d_in[19];
  float* out = (float*)d_out;

  const int N0 = in_sizes[0] / D, N1 = in_sizes[1] / D, N2 = in_sizes[2] / D;
  const int E0 = in_sizes[3], E1 = in_sizes[5], E2 = in_sizes[7];
  const size_t DD = (size_t)D * D;

  // workspace: 5 big [N2,128] buffers + 2 degree-scale vectors
  size_t nd = (size_t)N2 * D;
  float* B0 = (float*)d_ws;      // fusion input staging
  float* B1 = B0 + nd;           // hn / cat GEMM output
  float* B2 = B1 + nd;           // agg
  float* B3 = B2 + nd;           // resA (relu half)
  float* B4 = B3 + nd;           // resB (linear half)
  float* cntO = B4 + nd;         // rsqrt(deg_out)
  float* cntI = cntO + N2;       // rsqrt(deg_in)

  const size_t SH128 = (size_t)(128 * (128 + 4) + 128 * 128) * sizeof(float); // ~130 KB
  const size_t SH256 = (size_t)(64 * (256 + 4) + 256 * 128) * sizeof(float);  // ~193 KB

  auto degrees = [&](const int* s, const int* d, int n, int E) {
    hipMemsetAsync(cntO, 0, (size_t)n * sizeof(float), stream);
    hipMemsetAsync(cntI, 0, (size_t)n * sizeof(float), stream);
    deg_kernel<<<cd(E, 256), 256, 0, stream>>>(s, cntO, E);
    deg_kernel<<<cd(E, 256), 256, 0, stream>>>(d, cntI, E);
    scale_kernel<<<cd(n, 256), 256, 0, stream>>>(cntO, n);
    scale_kernel<<<cd(n, 256), 256, 0, stream>>>(cntI, n);
  };

  auto gconv_accum = [&](const float* X, const float* Wm, const float* bias, const float* sw,
                         const int* s, const int* d, int n, int E, int init) {
    gemm_wmma<128, 128><<<cd(n, 128), 256, SH128, stream>>>(X, nullptr, cntO, Wm, B1, n);
    hipMemsetAsync(B2, 0, (size_t)n * D * sizeof(float), stream);
    scatter_kernel<<<cd((long long)E * 32, 256), 256, 0, stream>>>(B1, s, d, B2, E);
    fuse_kernel<<<cd((long long)n * 32, 256), 256, 0, stream>>>(B2, B1, cntI, bias, sw, B3, B4, n, init);
  };

  auto finish = [&](int pi, int n, float* op) {
    gemm_wmma<256, 64><<<cd(n, 64), 128, SH256, stream>>>(
        B3, B4, nullptr, catW + (size_t)pi * 2 * DD, B1, n);
    ln_kernel<<<cd(n, 8), 256, 0, stream>>>(B1, catb + pi * D, lng + pi * D, lnb + pi * D, op, n);
  };

  // ---- level 0 (params 0): fusion = td0 = segment_sum(h1, dst0) ----
  hipMemsetAsync(B0, 0, (size_t)N0 * D * sizeof(float), stream);
  scatter_kernel<<<cd((long long)E0 * 32, 256), 256, 0, stream>>>(h1, nullptr, dst0, B0, E0);
  degrees(src0, dst0, N0, E0);
  gconv_accum(h0, convW, convb, convw, src0, dst0, N0, E0, 1);
  gconv_accum(B0, fusW, fusb, tdw, src0, dst0, N0, E0, 0);
  finish(0, N0, out);

  // ---- level 1 (params 2): fusion = bu1 = h0[dst0], td1 = segment_sum(h2, dst1) ----
  gather_kernel<<<cd((long long)N1 * 32, 256), 256, 0, stream>>>(h0, dst0, B0, N1); // N1 == E0
  degrees(src1, dst1, N1, E1);
  gconv_accum(h1, convW + 2 * DD, convb + 2 * D, convw + 2 * D, src1, dst1, N1, E1, 1);
  gconv_accum(B0, fusW + 2 * DD, fusb + 2 * D, buw + 2 * D, src1, dst1, N1, E1, 0);
  hipMemsetAsync(B0, 0, (size_t)N1 * D * sizeof(float), stream);
  scatter_kernel<<<cd((long long)E1 * 32, 256), 256, 0, stream>>>(h2, nullptr, dst1, B0, E1);
  gconv_accum(B0, fusW + 2 * DD, fusb + 2 * D, tdw + 2 * D, src1, dst1, N1, E1, 0);
  finish(2, N1, out + (size_t)N0 * D);

  // ---- level 2 (params 1): fusion = bu2 = h1[dst1] ----
  gather_kernel<<<cd((long long)N2 * 32, 256), 256, 0, stream>>>(h1, dst1, B0, N2); // N2 == E1
  degrees(src2, dst2, N2, E2);
  gconv_accum(h2, convW + DD, convb + D, convw + D, src2, dst2, N2, E2, 1);
  gconv_accum(B0, fusW + DD, fusb + D, buw + D, src2, dst2, N2, E2, 0);
  finish(1, N2, out + (size_t)(N0 + N1) * D);
}